// CopulaDecoder_87668872446459
// MI455X (gfx1250) — compile-verified
//
#include <hip/hip_runtime.h>
#include <hip/hip_bf16.h>
#include <math.h>

typedef __attribute__((ext_vector_type(2))) float v2f;
typedef __attribute__((ext_vector_type(4))) float v4f;
typedef __attribute__((ext_vector_type(8))) float v8f;

#define VN 16
#define TN 2048
#define NN (VN * TN)     // 32768
#define DN 64
#define HN 128
#define DKN 48

#define XS_STRIDE 68     // 64-col tile, padded (bank-conflict free, 8B aligned)
#define AS_STRIDE 116    // 112-col tile, padded
#define HS_STRIDE 132    // 128-col tile, padded

__device__ __forceinline__ v8f wmma_f32(v2f a, v2f b, v8f c) {
    // V_WMMA_F32_16X16X4_F32 : D = A(16x4) x B(4x16) + C(16x16)
    return __builtin_amdgcn_wmma_f32_16x16x4_f32(false, a, false, b, (short)0, c, false, false);
}

// ---------------------------------------------------------------------------
// Kernel 0: jittered max_depth via JAX threefry2x32 (partitionable counter
// mode): per element i, x=(0,i), key=(0,42), bits = out0 ^ out1,
// u = max(0, bitcast(bits>>9 | 0x3f800000) - 1),  mdj = max_depth + u/(4T).
// ---------------------------------------------------------------------------
__global__ __launch_bounds__(256) void jitter_kernel(const float* __restrict__ md_in,
                                                     float* __restrict__ mdj) {
    int i = blockIdx.x * blockDim.x + threadIdx.x;
    if (i >= NN) return;
    unsigned k0 = 0u, k1 = 42u;
    unsigned k2 = 0x1BD11BDAu ^ k0 ^ k1;
    unsigned x0 = 0u, x1 = (unsigned)i;
    x0 += k0; x1 += k1;
#define TFR(r) { x0 += x1; x1 = (x1 << (r)) | (x1 >> (32 - (r))); x1 ^= x0; }
    TFR(13) TFR(15) TFR(26) TFR(6)   x0 += k1; x1 += k2 + 1u;
    TFR(17) TFR(29) TFR(16) TFR(24)  x0 += k2; x1 += k0 + 2u;
    TFR(13) TFR(15) TFR(26) TFR(6)   x0 += k0; x1 += k1 + 3u;
    TFR(17) TFR(29) TFR(16) TFR(24)  x0 += k1; x1 += k2 + 4u;
    TFR(13) TFR(15) TFR(26) TFR(6)   x0 += k2; x1 += k0 + 5u;
#undef TFR
    unsigned bits = x0 ^ x1;
    float uf = __uint_as_float((bits >> 9) | 0x3f800000u) - 1.0f;
    uf = fmaxf(uf, 0.0f);
    mdj[i] = md_in[i] + uf * (1.0f / 8192.0f);   // /(4*T)
}

// ---------------------------------------------------------------------------
// Kernel 1: Q/K/V MLPs.  Per wave: 16 rows.  X tile in LDS, GEMM1 (64->128)
// via WMMA f32 16x16x4, ReLU -> LDS, GEMM2 (128->48) via WMMA.
// Block = 128 threads (4 waves) -> 64 rows/block, grid = 512.
// ---------------------------------------------------------------------------
__global__ __launch_bounds__(128) void qkv_kernel(
    const float* __restrict__ enc,
    const float* __restrict__ Wq1, const float* __restrict__ bq1,
    const float* __restrict__ Wq2, const float* __restrict__ bq2,
    const float* __restrict__ Wk1, const float* __restrict__ bk1,
    const float* __restrict__ Wk2, const float* __restrict__ bk2,
    const float* __restrict__ Wv1, const float* __restrict__ bv1,
    const float* __restrict__ Wv2, const float* __restrict__ bv2,
    float* __restrict__ Qo, float* __restrict__ Ko, float* __restrict__ Vo) {
    __shared__ float xs[4][16 * XS_STRIDE];
    __shared__ float hs[4][16 * HS_STRIDE];
    const int w = threadIdx.x >> 5, lane = threadIdx.x & 31;
    const int nlo = lane & 15, hi = lane >> 4;
    const int row0 = blockIdx.x * 64 + w * 16;

    for (int i = lane; i < 16 * 64; i += 32) {
        int r = i >> 6, c = i & 63;
        xs[w][r * XS_STRIDE + c] = enc[(row0 + r) * DN + c];
    }
    __syncthreads();

    const float* W1s[3] = {Wq1, Wk1, Wv1};
    const float* b1s[3] = {bq1, bk1, bv1};
    const float* W2s[3] = {Wq2, Wk2, Wv2};
    const float* b2s[3] = {bq2, bk2, bv2};
    float* Os[3] = {Qo, Ko, Vo};

#pragma unroll
    for (int s = 0; s < 3; ++s) {
        const float* W1 = W1s[s];
        const float* W2 = W2s[s];
        // ---- GEMM1: [16,64] @ [64,128] + b1, ReLU ----
        v8f acc[8];
#pragma unroll
        for (int ct = 0; ct < 8; ++ct) {
            float bv = b1s[s][ct * 16 + nlo];
#pragma unroll
            for (int r = 0; r < 8; ++r) acc[ct][r] = bv;
        }
        for (int kk = 0; kk < 16; ++kk) {
            int kb = kk * 4 + hi * 2;
            v2f a = *(const v2f*)&xs[w][nlo * XS_STRIDE + kb];
#pragma unroll
            for (int ct = 0; ct < 8; ++ct) {
                v2f b;
                b.x = W1[kb * HN + ct * 16 + nlo];
                b.y = W1[(kb + 1) * HN + ct * 16 + nlo];
                acc[ct] = wmma_f32(a, b, acc[ct]);
            }
        }
#pragma unroll
        for (int ct = 0; ct < 8; ++ct)
#pragma unroll
            for (int r = 0; r < 8; ++r)
                hs[w][(hi * 8 + r) * HS_STRIDE + ct * 16 + nlo] = fmaxf(acc[ct][r], 0.0f);
        __syncthreads();
        // ---- GEMM2: [16,128] @ [128,48] + b2 ----
        v8f acc2[3];
#pragma unroll
        for (int ct = 0; ct < 3; ++ct) {
            float bv = b2s[s][ct * 16 + nlo];
#pragma unroll
            for (int r = 0; r < 8; ++r) acc2[ct][r] = bv;
        }
        for (int kk = 0; kk < 32; ++kk) {
            int kb = kk * 4 + hi * 2;
            v2f a = *(const v2f*)&hs[w][nlo * HS_STRIDE + kb];
#pragma unroll
            for (int ct = 0; ct < 3; ++ct) {
                v2f b;
                b.x = W2[kb * DKN + ct * 16 + nlo];
                b.y = W2[(kb + 1) * DKN + ct * 16 + nlo];
                acc2[ct] = wmma_f32(a, b, acc2[ct]);
            }
        }
        float* O = Os[s];
#pragma unroll
        for (int ct = 0; ct < 3; ++ct)
#pragma unroll
            for (int r = 0; r < 8; ++r)
                O[(row0 + hi * 8 + r) * DKN + ct * 16 + nlo] = acc2[ct][r];
        __syncthreads();
    }
}

// ---------------------------------------------------------------------------
// Kernel 2: gather-attention.  One wave per query n; lane j = neighbor slot
// (lane 0 also handles slot 32 = window).  Wave softmax, ctx -> workspace.
// ---------------------------------------------------------------------------
__device__ __forceinline__ float dot48(const v4f qv[12], const float* __restrict__ krow,
                                       const float* __restrict__ rrow) {
    float s = 0.0f;
#pragma unroll
    for (int i = 0; i < 12; ++i) {
        v4f kv = ((const v4f*)krow)[i];
        v4f rv = ((const v4f*)rrow)[i];
        s = fmaf(qv[i].x, kv.x + rv.x, s);
        s = fmaf(qv[i].y, kv.y + rv.y, s);
        s = fmaf(qv[i].z, kv.z + rv.z, s);
        s = fmaf(qv[i].w, kv.w + rv.w, s);
    }
    return s;
}

__global__ __launch_bounds__(256) void attn_kernel(
    const float* __restrict__ Q, const float* __restrict__ Km, const float* __restrict__ Vm,
    const float* __restrict__ role_emb,
    const int* __restrict__ prev_values, const int* __restrict__ next_values,
    const int* __restrict__ window_indices,
    const float* __restrict__ prev_mask, const float* __restrict__ next_mask,
    const float* __restrict__ window_mask,
    const float* __restrict__ mdj, float* __restrict__ CTX) {
    __shared__ float s_attn[8][34];
    __shared__ int s_nbr[8][34];
    const int w = threadIdx.x >> 5, lane = threadIdx.x & 31;
    const int n = blockIdx.x * 8 + w;
    const int vq = n >> 11, t = n & (TN - 1);

    v4f qv[12];
    {
        const v4f* qp = (const v4f*)(Q + (long)n * DKN);
#pragma unroll
        for (int i = 0; i < 12; ++i) qv[i] = qp[i];
    }
    const float mdq = mdj[n];
    int nbrj, role;
    bool valid;
    if (lane < 16) {
        int vk = lane, idx = vk * TN + t;
        nbrj = prev_values[idx];
        role = 16 * vq + vk;
        valid = (prev_mask[idx] == 0.0f);
    } else {
        int vk = lane - 16, idx = vk * TN + t;
        bool comp = mdq > mdj[idx];
        nbrj = comp ? idx : next_values[idx];
        role = 256 + 16 * vq + vk;
        valid = comp || (next_mask[idx] == 0.0f);
    }
    const float inv = 0.14433756729740643f;  // 1/sqrt(48)
    float s = dot48(qv, Km + (long)nbrj * DKN, role_emb + (long)role * DKN);
    s = valid ? s * inv : -1e9f;
    float s2 = -1e9f;
    int nbr2 = 0;
    if (lane == 0) {
        nbr2 = window_indices[n];
        float sw = dot48(qv, Km + (long)nbr2 * DKN, role_emb + (long)(512 + vq) * DKN);
        s2 = (window_mask[n] == 0.0f) ? sw * inv : -1e9f;
    }
    float m = fmaxf(s, s2);
#pragma unroll
    for (int off = 16; off > 0; off >>= 1) m = fmaxf(m, __shfl_xor(m, off, 32));
    float e = expf(s - m);
    float e2 = (lane == 0) ? expf(s2 - m) : 0.0f;
    float sum = e + e2;
#pragma unroll
    for (int off = 16; off > 0; off >>= 1) sum += __shfl_xor(sum, off, 32);
    s_attn[w][lane] = e / sum;
    s_nbr[w][lane] = nbrj;
    if (lane == 0) { s_attn[w][32] = e2 / sum; s_nbr[w][32] = nbr2; }
    __syncthreads();
    float c0 = 0.0f, c1 = 0.0f;
    for (int j = 0; j < 33; ++j) {
        float a = s_attn[w][j];
        int nb = s_nbr[w][j];
        c0 = fmaf(a, Vm[(long)nb * DKN + lane], c0);
        if (lane < 16) c1 = fmaf(a, Vm[(long)nb * DKN + 32 + lane], c1);
    }
    CTX[(long)n * DKN + lane] = c0;
    if (lane < 16) CTX[(long)n * DKN + 32 + lane] = c1;
}

// ---------------------------------------------------------------------------
// Kernel 3: dp = MLP([enc,ctx]) (112->128->2), mp = MLP(enc) (64->128->2),
// then Gaussian NLL.  Same WMMA tiling; 16 rows/wave, 4 waves/block.
// ---------------------------------------------------------------------------
__device__ __forceinline__ float softplusf(float x) {
    return fmaxf(x, 0.0f) + log1pf(expf(-fabsf(x)));
}
__device__ __forceinline__ float nlogp(float x, float mu, float ls) {
    float sp = softplusf(ls) + 1e-3f;
    float z = (x - mu) / sp;
    return 0.5f * z * z + logf(sp) + 0.9189385332046727f;
}

__global__ __launch_bounds__(128) void head_kernel(
    const float* __restrict__ enc, const float* __restrict__ CTX, const float* __restrict__ u,
    const float* __restrict__ Wd1, const float* __restrict__ bd1,
    const float* __restrict__ Wd2, const float* __restrict__ bd2,
    const float* __restrict__ Wm1, const float* __restrict__ bm1,
    const float* __restrict__ Wm2, const float* __restrict__ bm2,
    float* __restrict__ out) {
    __shared__ float as_[4][16 * AS_STRIDE];
    __shared__ float hs[4][16 * HS_STRIDE];
    __shared__ float dpb[4][32];
    __shared__ float mpb[4][32];
    const int w = threadIdx.x >> 5, lane = threadIdx.x & 31;
    const int nlo = lane & 15, hi = lane >> 4;
    const int row0 = blockIdx.x * 64 + w * 16;

    for (int i = lane; i < 16 * 64; i += 32) {
        int r = i >> 6, c = i & 63;
        as_[w][r * AS_STRIDE + c] = enc[(row0 + r) * DN + c];
    }
    for (int i = lane; i < 16 * 48; i += 32) {
        int r = i / 48, c = i - r * 48;
        as_[w][r * AS_STRIDE + 64 + c] = CTX[(row0 + r) * DKN + c];
    }
    __syncthreads();

    v8f acc[8];
    // ---- dp layer 1: [16,112] @ [112,128], ReLU ----
#pragma unroll
    for (int ct = 0; ct < 8; ++ct) {
        float bv = bd1[ct * 16 + nlo];
#pragma unroll
        for (int r = 0; r < 8; ++r) acc[ct][r] = bv;
    }
    for (int kk = 0; kk < 28; ++kk) {
        int kb = kk * 4 + hi * 2;
        v2f a = *(const v2f*)&as_[w][nlo * AS_STRIDE + kb];
#pragma unroll
        for (int ct = 0; ct < 8; ++ct) {
            v2f b;
            b.x = Wd1[kb * HN + ct * 16 + nlo];
            b.y = Wd1[(kb + 1) * HN + ct * 16 + nlo];
            acc[ct] = wmma_f32(a, b, acc[ct]);
        }
    }
#pragma unroll
    for (int ct = 0; ct < 8; ++ct)
#pragma unroll
        for (int r = 0; r < 8; ++r)
            hs[w][(hi * 8 + r) * HS_STRIDE + ct * 16 + nlo] = fmaxf(acc[ct][r], 0.0f);
    __syncthreads();
    // ---- dp layer 2: [16,128] @ [128,2] ----
    v8f a2;
    {
        float bv = (nlo < 2) ? bd2[nlo] : 0.0f;
#pragma unroll
        for (int r = 0; r < 8; ++r) a2[r] = bv;
    }
    for (int kk = 0; kk < 32; ++kk) {
        int kb = kk * 4 + hi * 2;
        v2f a = *(const v2f*)&hs[w][nlo * HS_STRIDE + kb];
        v2f b;
        b.x = (nlo < 2) ? Wd2[kb * 2 + nlo] : 0.0f;
        b.y = (nlo < 2) ? Wd2[(kb + 1) * 2 + nlo] : 0.0f;
        a2 = wmma_f32(a, b, a2);
    }
    if (nlo < 2)
#pragma unroll
        for (int r = 0; r < 8; ++r) dpb[w][(hi * 8 + r) * 2 + nlo] = a2[r];
    __syncthreads();
    // ---- mp layer 1: [16,64] @ [64,128], ReLU ----
#pragma unroll
    for (int ct = 0; ct < 8; ++ct) {
        float bv = bm1[ct * 16 + nlo];
#pragma unroll
        for (int r = 0; r < 8; ++r) acc[ct][r] = bv;
    }
    for (int kk = 0; kk < 16; ++kk) {
        int kb = kk * 4 + hi * 2;
        v2f a = *(const v2f*)&as_[w][nlo * AS_STRIDE + kb];
#pragma unroll
        for (int ct = 0; ct < 8; ++ct) {
            v2f b;
            b.x = Wm1[kb * HN + ct * 16 + nlo];
            b.y = Wm1[(kb + 1) * HN + ct * 16 + nlo];
            acc[ct] = wmma_f32(a, b, acc[ct]);
        }
    }
#pragma unroll
    for (int ct = 0; ct < 8; ++ct)
#pragma unroll
        for (int r = 0; r < 8; ++r)
            hs[w][(hi * 8 + r) * HS_STRIDE + ct * 16 + nlo] = fmaxf(acc[ct][r], 0.0f);
    __syncthreads();
    // ---- mp layer 2 ----
    {
        float bv = (nlo < 2) ? bm2[nlo] : 0.0f;
#pragma unroll
        for (int r = 0; r < 8; ++r) a2[r] = bv;
    }
    for (int kk = 0; kk < 32; ++kk) {
        int kb = kk * 4 + hi * 2;
        v2f a = *(const v2f*)&hs[w][nlo * HS_STRIDE + kb];
        v2f b;
        b.x = (nlo < 2) ? Wm2[kb * 2 + nlo] : 0.0f;
        b.y = (nlo < 2) ? Wm2[(kb + 1) * 2 + nlo] : 0.0f;
        a2 = wmma_f32(a, b, a2);
    }
    if (nlo < 2)
#pragma unroll
        for (int r = 0; r < 8; ++r) mpb[w][(hi * 8 + r) * 2 + nlo] = a2[r];
    __syncthreads();
    // ---- NLL ----
    if (lane < 16) {
        int n = row0 + lane;
        float uv = u[n];
        float dmu = dpb[w][lane * 2 + 0], dls = dpb[w][lane * 2 + 1];
        float mmu = mpb[w][lane * 2 + 0], mls = mpb[w][lane * 2 + 1];
        out[n] = nlogp(uv, dmu, dls) + nlogp(uv, mmu, mls);
    }
}

// ---------------------------------------------------------------------------
extern "C" void kernel_launch(void* const* d_in, const int* in_sizes, int n_in,
                              void* d_out, int out_size, void* d_ws, size_t ws_size,
                              hipStream_t stream) {
    (void)in_sizes; (void)n_in; (void)out_size; (void)ws_size;
    const float* enc  = (const float*)d_in[0];
    const float* u    = (const float*)d_in[1];
    const float* Wq1  = (const float*)d_in[2];
    const float* bq1  = (const float*)d_in[3];
    const float* Wq2  = (const float*)d_in[4];
    const float* bq2  = (const float*)d_in[5];
    const float* Wk1  = (const float*)d_in[6];
    const float* bk1  = (const float*)d_in[7];
    const float* Wk2  = (const float*)d_in[8];
    const float* bk2  = (const float*)d_in[9];
    const float* Wv1  = (const float*)d_in[10];
    const float* bv1  = (const float*)d_in[11];
    const float* Wv2  = (const float*)d_in[12];
    const float* bv2  = (const float*)d_in[13];
    const float* role = (const float*)d_in[14];
    const float* Wd1  = (const float*)d_in[15];
    const float* bd1  = (const float*)d_in[16];
    const float* Wd2  = (const float*)d_in[17];
    const float* bd2  = (const float*)d_in[18];
    const float* Wm1  = (const float*)d_in[19];
    const float* bm1  = (const float*)d_in[20];
    const float* Wm2  = (const float*)d_in[21];
    const float* bm2  = (const float*)d_in[22];
    const int* prev_values    = (const int*)d_in[23];
    const int* next_values    = (const int*)d_in[24];
    const int* window_indices = (const int*)d_in[25];
    // d_in[26] = indices (recomputed on the fly)
    const float* prev_mask   = (const float*)d_in[27];
    const float* next_mask   = (const float*)d_in[28];
    const float* window_mask = (const float*)d_in[29];
    const float* max_depth   = (const float*)d_in[30];

    float* ws  = (float*)d_ws;
    float* mdj = ws;                       // 32768
    float* Q   = mdj + NN;                 // 32768*48
    float* Km  = Q + (size_t)NN * DKN;
    float* Vm  = Km + (size_t)NN * DKN;
    float* CTX = Vm + (size_t)NN * DKN;

    jitter_kernel<<<(NN + 255) / 256, 256, 0, stream>>>(max_depth, mdj);
    qkv_kernel<<<NN / 64, 128, 0, stream>>>(enc, Wq1, bq1, Wq2, bq2, Wk1, bk1, Wk2, bk2,
                                            Wv1, bv1, Wv2, bv2, Q, Km, Vm);
    attn_kernel<<<NN / 8, 256, 0, stream>>>(Q, Km, Vm, role, prev_values, next_values,
                                            window_indices, prev_mask, next_mask,
                                            window_mask, mdj, CTX);
    head_kernel<<<NN / 64, 128, 0, stream>>>(enc, CTX, u, Wd1, bd1, Wd2, bd2,
                                             Wm1, bm1, Wm2, bm2, (float*)d_out);
}